// SoftOC_Module_46385646797212
// MI455X (gfx1250) — compile-verified
//
#include <hip/hip_runtime.h>
#include <math.h>

typedef __attribute__((ext_vector_type(2))) float v2f;
typedef __attribute__((ext_vector_type(8))) float v8f;

#define NB   8
#define KC   19          // classes
#define CF   512         // feature channels
#define HW   9216        // 96*96
#define KPAD 20          // classes padded to multiple of 4

// ---- workspace layout (floats) ----
#define OFF_P     0
#define SZ_P      (NB * KC * HW)        // softmax probs: 1,400,832 f (5.6 MB)
#define OFF_DEN   (OFF_P + SZ_P)
#define SZ_DEN    32
#define OFF_POOL  (OFF_DEN + SZ_DEN)
#define SZ_POOL   (NB * KC * CF)        // 77,824 f
#define OFF_CTX   (OFF_POOL + SZ_POOL)
#define SZ_CTX    (NB * CF * KPAD)      // 81,920 f  (total ~6.25 MB)

// -------------------- K0: zero the global denominator --------------------
__global__ void zero_denom_kernel(float* __restrict__ denom) {
    if (threadIdx.x < SZ_DEN) denom[threadIdx.x] = 0.0f;
}

// -------------------- K1: softmax over classes + denom reduction --------------------
// one thread per (n, hw) pixel; grid = N*HW/256 exactly
__global__ void softmax_kernel(const float* __restrict__ probs,
                               float* __restrict__ p,
                               float* __restrict__ denom) {
    __shared__ float sden[KC];
    const int t = threadIdx.x;
    if (t < KC) sden[t] = 0.0f;
    __syncthreads();

    const int idx = blockIdx.x * blockDim.x + t;   // n*HW + hw  (exact grid)
    const int n  = idx / HW;
    const int hw = idx % HW;

    const float* __restrict__ base = probs + (size_t)n * KC * HW + hw;
    float v[KC];
    float m = -INFINITY;
#pragma unroll
    for (int k = 0; k < KC; ++k) { v[k] = base[(size_t)k * HW]; m = fmaxf(m, v[k]); }
    float s = 0.0f;
#pragma unroll
    for (int k = 0; k < KC; ++k) { v[k] = __expf(v[k] - m); s += v[k]; }
    const float inv = 1.0f / s;

    float* __restrict__ pb = p + (size_t)n * KC * HW + hw;
#pragma unroll
    for (int k = 0; k < KC; ++k) {
        const float pk = v[k] * inv;
        pb[(size_t)k * HW] = pk;
        atomicAdd(&sden[k], pk);           // ds_add_f32
    }
    __syncthreads();
    if (t < KC) atomicAdd(&denom[t], sden[t]);   // global_atomic_add_f32
}

// -------------------- K2: pooled[n,k,c] = sum_hw p[n,k,hw]*feats[n,c,hw] --------------------
// GEMM per n: M=classes (2 tiles of 16, padded), N=channels (tile of 16), Kdim=HW.
// Block = 256 threads = 8 waves; grid = (CF/16, NB). Waves split HW, reduce in LDS.
__global__ void pooled_kernel(const float* __restrict__ feats,
                              const float* __restrict__ p,
                              float* __restrict__ pooled) {
    __shared__ float red[8][2][32][8];     // [wave][mtile][lane][reg] = 16 KB

    const int n    = blockIdx.y;
    const int c0   = blockIdx.x * 16;
    const int wv   = threadIdx.x >> 5;
    const int lane = threadIdx.x & 31;
    const int lo   = lane & 15;
    const int hi   = lane >> 4;

    const float* __restrict__ fb  = feats + (size_t)n * CF * HW + (size_t)(c0 + lo) * HW;
    const float* __restrict__ pa0 = p + (size_t)n * KC * HW + (size_t)lo * HW;
    const int  r1  = (lo < 3) ? (16 + lo) : 0;             // clamped row for m-tile 1
    const bool a1v = (lo < 3);
    const float* __restrict__ pa1 = p + (size_t)n * KC * HW + (size_t)r1 * HW;

    v8f acc0 = {};
    v8f acc1 = {};

    const int hwbase = wv * (HW / 8) + 2 * hi;             // per-wave K slice
#pragma unroll 4
    for (int it = 0; it < (HW / 8) / 4; ++it) {
        const int hv = hwbase + it * 4;
        const v2f b  = *(const v2f*)(fb  + hv);            // feats tile (shared by both WMMAs)
        const v2f a0 = *(const v2f*)(pa0 + hv);            // classes 0..15
        const v2f at = *(const v2f*)(pa1 + hv);            // classes 16..18 (clamped)
        v2f a1; a1.x = a1v ? at.x : 0.0f; a1.y = a1v ? at.y : 0.0f;
        acc0 = __builtin_amdgcn_wmma_f32_16x16x4_f32(false, a0, false, b, (short)0, acc0, false, false);
        acc1 = __builtin_amdgcn_wmma_f32_16x16x4_f32(false, a1, false, b, (short)0, acc1, false, false);
    }

#pragma unroll
    for (int r = 0; r < 8; ++r) { red[wv][0][lane][r] = acc0[r]; red[wv][1][lane][r] = acc1[r]; }
    __syncthreads();

    if (wv < 2) {  // wave 0 reduces m-tile 0, wave 1 reduces m-tile 1
#pragma unroll
        for (int r = 0; r < 8; ++r) {
            float s = 0.0f;
#pragma unroll
            for (int ww = 0; ww < 8; ++ww) s += red[ww][wv][lane][r];
            const int m = wv * 16 + r + 8 * hi;            // class index
            if (m < KC)
                pooled[(size_t)n * KC * CF + (size_t)m * CF + c0 + lo] = s;
        }
    }
}

// -------------------- K3: ctxT[n][c][kp] = pooled[n][kp][c] / denom[kp], zero-padded --------------------
__global__ void context_kernel(const float* __restrict__ pooled,
                               const float* __restrict__ denom,
                               float* __restrict__ ctxT) {
    const int idx = blockIdx.x * blockDim.x + threadIdx.x;
    if (idx >= NB * CF * KPAD) return;
    const int kp = idx % KPAD;
    const int c  = (idx / KPAD) % CF;
    const int n  = idx / (KPAD * CF);
    float v = 0.0f;
    if (kp < KC) v = pooled[(size_t)n * KC * CF + (size_t)kp * CF + c] / denom[kp];
    ctxT[idx] = v;
}

// -------------------- K4: out[n,c,hw] = sum_k ctxT[n,c,k] * p[n,k,hw] --------------------
// GEMM per n: M=channels (32 tiles), N=hw (576 tiles), Kdim=KPAD=20 -> 5 WMMAs.
// Block = 256 threads = 8 waves; each wave owns one 16x16 out tile.
// grid = (HW/16/8, CF/16, NB)
__global__ void out_kernel(const float* __restrict__ ctxT,
                           const float* __restrict__ p,
                           float* __restrict__ out) {
    const int n    = blockIdx.z;
    const int c0   = blockIdx.y * 16;
    const int wv   = threadIdx.x >> 5;
    const int lane = threadIdx.x & 31;
    const int lo   = lane & 15;
    const int hi   = lane >> 4;
    const int hw0  = (blockIdx.x * 8 + wv) * 16;

    const float* __restrict__ ab = ctxT + (size_t)n * CF * KPAD + (size_t)(c0 + lo) * KPAD;
    const float* __restrict__ pb = p + (size_t)n * KC * HW + hw0 + lo;

    v8f acc = {};
#pragma unroll
    for (int kc = 0; kc < 5; ++kc) {
        const v2f a = *(const v2f*)(ab + kc * 4 + 2 * hi);   // padded row -> always in range
        const int k0 = kc * 4 + 2 * hi;                      // <= 18 always
        v2f b;
        b.x = pb[(size_t)k0 * HW];
        b.y = (k0 + 1 < KC) ? pb[(size_t)(k0 + 1) * HW] : 0.0f;  // only kc==4,hi==1 masks
        acc = __builtin_amdgcn_wmma_f32_16x16x4_f32(false, a, false, b, (short)0, acc, false, false);
    }

    float* __restrict__ ob = out + (size_t)n * CF * HW + (size_t)c0 * HW + hw0 + lo;
#pragma unroll
    for (int r = 0; r < 8; ++r)
        ob[(size_t)(r + 8 * hi) * HW] = acc[r];
}

extern "C" void kernel_launch(void* const* d_in, const int* in_sizes, int n_in,
                              void* d_out, int out_size, void* d_ws, size_t ws_size,
                              hipStream_t stream) {
    (void)in_sizes; (void)n_in; (void)out_size; (void)ws_size;

    const float* feats = (const float*)d_in[0];   // (8, 512, 96, 96)
    const float* probs = (const float*)d_in[1];   // (8, 19, 96, 96)
    float* out = (float*)d_out;                   // (8, 512, 96, 96)

    float* ws     = (float*)d_ws;
    float* p      = ws + OFF_P;
    float* denom  = ws + OFF_DEN;
    float* pooled = ws + OFF_POOL;
    float* ctxT   = ws + OFF_CTX;

    zero_denom_kernel<<<dim3(1), dim3(32), 0, stream>>>(denom);
    softmax_kernel<<<dim3((NB * HW) / 256), dim3(256), 0, stream>>>(probs, p, denom);
    pooled_kernel<<<dim3(CF / 16, NB), dim3(256), 0, stream>>>(feats, p, pooled);
    context_kernel<<<dim3((NB * CF * KPAD + 255) / 256), dim3(256), 0, stream>>>(pooled, denom, ctxT);
    out_kernel<<<dim3(HW / 16 / 8, CF / 16, NB), dim3(256), 0, stream>>>(ctxT, p, out);
}